// CEFM_47863115547253
// MI455X (gfx1250) — compile-verified
//
#include <hip/hip_runtime.h>
#include <math.h>

#define B_   16
#define TF_  8192
#define TP_  512
#define TF2_ 4096

typedef __attribute__((ext_vector_type(2))) float v2f;
typedef __attribute__((ext_vector_type(8))) float v8f;

// ---------------------------------------------------------------------------
// Stage 1a: prev/next voiced-index scans. One wave32 per batch row.
// prev[t] = cummax(where(voiced, t, -1));  next[t] = rev-cummin(where(voiced, t, TF))
// ---------------------------------------------------------------------------
__global__ __launch_bounds__(32) void scan_kernel(const float* __restrict__ f0,
                                                  int* __restrict__ prevA,
                                                  int* __restrict__ nextA) {
  const int b = blockIdx.x;
  const int lane = threadIdx.x;
  const float* row = f0 + (size_t)b * TF_;
  int* pr = prevA + (size_t)b * TF_;
  int* nx = nextA + (size_t)b * TF_;
  const int C = TF_ / 32;           // 256 elements per lane (contiguous chunk)
  const int t0 = lane * C;

  // ---- prev: forward cummax of voiced index ----
  int m = -1;
  for (int i = 0; i < C; ++i) if (row[t0 + i] > 0.0f) m = t0 + i;
  int inc = m;
#pragma unroll
  for (int off = 1; off < 32; off <<= 1) {
    int u = __shfl_up(inc, off, 32);
    if (lane >= off) inc = (inc > u) ? inc : u;
  }
  int carry = __shfl_up(inc, 1, 32);
  if (lane == 0) carry = -1;        // exclusive carry into this chunk
  for (int i = 0; i < C; ++i) {
    int t = t0 + i;
    if (row[t] > 0.0f) carry = t;
    pr[t] = carry;
  }

  // ---- next: reverse cummin of voiced index ----
  int m2 = TF_;
  for (int i = C - 1; i >= 0; --i) if (row[t0 + i] > 0.0f) m2 = t0 + i;
  int inc2 = m2;
#pragma unroll
  for (int off = 1; off < 32; off <<= 1) {
    int u = __shfl_down(inc2, off, 32);
    if (lane + off < 32) inc2 = (inc2 < u) ? inc2 : u;
  }
  int carry2 = __shfl_down(inc2, 1, 32);
  if (lane == 31) carry2 = TF_;
  for (int i = C - 1; i >= 0; --i) {
    int t = t0 + i;
    if (row[t] > 0.0f) carry2 = t;
    nx[t] = carry2;
  }
}

// ---------------------------------------------------------------------------
// Stage 1b: linear interpolation of unvoiced frames (with edge clamping).
// ---------------------------------------------------------------------------
__global__ __launch_bounds__(256) void interp_kernel(const float* __restrict__ f0,
                                                     const int* __restrict__ prevA,
                                                     const int* __restrict__ nextA,
                                                     float* __restrict__ interp) {
  int idx = blockIdx.x * 256 + threadIdx.x;   // over B*TF
  int b = idx / TF_;
  int t = idx - b * TF_;
  const float* row = f0 + (size_t)b * TF_;
  float fv = row[t];
  float out;
  if (fv > 0.0f) {
    out = fv;
  } else {
    int p = prevA[idx], n = nextA[idx];
    int lo = (p >= 0) ? p : n;      // before first voiced -> clamp to first
    int hi = (n < TF_) ? n : p;     // after last voiced -> clamp to last
    float lov = row[lo], hiv = row[hi];
    float span = fmaxf((float)(hi - lo), 1.0f);
    float w = fminf(fmaxf(((float)(t - lo)) / span, 0.0f), 1.0f);
    out = lov + w * (hiv - lov);
  }
  interp[idx] = out;
}

// ---------------------------------------------------------------------------
// Stage 1c: SG(7,2) smooth (replicate pad) -> |d2| (replicate pad), then
// pairwise downsample by 2 (the nearest-upsampled attention uses each native
// frame exactly twice). Also pairwise-sum raw f0 for the mean einsum.
// ---------------------------------------------------------------------------
__device__ __forceinline__ float sg_at(const float* __restrict__ x, int t) {
  const float c0 = -2.0f / 21.0f, c1 = 3.0f / 21.0f, c2 = 6.0f / 21.0f, c3 = 7.0f / 21.0f;
  int tm3 = t - 3 > 0 ? t - 3 : 0, tm2 = t - 2 > 0 ? t - 2 : 0, tm1 = t - 1 > 0 ? t - 1 : 0;
  int tp3 = t + 3 < TF_ - 1 ? t + 3 : TF_ - 1, tp2 = t + 2 < TF_ - 1 ? t + 2 : TF_ - 1,
      tp1 = t + 1 < TF_ - 1 ? t + 1 : TF_ - 1;
  return c0 * (x[tm3] + x[tp3]) + c1 * (x[tm2] + x[tp2]) + c2 * (x[tm1] + x[tp1]) + c3 * x[t];
}
__device__ __forceinline__ float d2_at(const float* __restrict__ x, int t) {
  int tl = t - 1 > 0 ? t - 1 : 0;
  int tr = t + 1 < TF_ - 1 ? t + 1 : TF_ - 1;
  float l = sg_at(x, tl), c = sg_at(x, t), r = sg_at(x, tr);
  return fabsf(l - 2.0f * c + r);
}
__global__ __launch_bounds__(256) void pairsum_kernel(const float* __restrict__ f0,
                                                      const float* __restrict__ interp,
                                                      float* __restrict__ curv2,
                                                      float* __restrict__ f02) {
  int idx = blockIdx.x * 256 + threadIdx.x;   // over B*TF2
  int b = idx / TF2_;
  int t2 = idx - b * TF2_;
  const float* xr = interp + (size_t)b * TF_;
  const float* fr = f0 + (size_t)b * TF_;
  curv2[idx] = d2_at(xr, 2 * t2) + d2_at(xr, 2 * t2 + 1);
  f02[idx]   = fr[2 * t2] + fr[2 * t2 + 1];
}

// ---------------------------------------------------------------------------
// Stage 2: the bandwidth-dominant einsums as WMMA F32 16x16x4.
// One wave32 per (batch, 16-phoneme tile). A = attn tile (M=16 phonemes,
// K=4 native frames), B = 4x16 with col0=curv2, col1=f02, col2=2.0 (denom).
// Branchless B-fragment build (per-lane column weights are loop-invariant);
// 1-deep software pipeline: chunk k+1 global loads issued before the 16-WMMA
// burst on chunk k, so VMEM latency overlaps the matrix ops.
// ---------------------------------------------------------------------------
#define KC   64
#define TSTR 68   // padded LDS row stride (floats) to spread banks

__global__ __launch_bounds__(32) void einsum_wmma_kernel(const float* __restrict__ attn,
                                                         const float* __restrict__ curv2,
                                                         const float* __restrict__ f02,
                                                         float* __restrict__ out_curv,
                                                         float* __restrict__ out_mean) {
  __shared__ float sT[16 * TSTR];
  __shared__ float sC[KC];
  __shared__ float sF[KC];
  const int b = blockIdx.y;
  const int p_base = blockIdx.x * 16;
  const int lane = threadIdx.x;
  const float* ab = attn + ((size_t)b * TP_ + p_base) * TF2_;
  const float* cb = curv2 + (size_t)b * TF2_;
  const float* fb = f02 + (size_t)b * TF2_;

  const int n = lane & 15;           // B/C/D column held by this lane
  const int koff = (lane >> 4) * 2;  // K sub-offset for upper lane half (A & B layouts)
  const int arow = lane & 15;        // A row (phoneme within tile)
  const int rhalf = lane >> 4;       // tile-load row parity
  const int col4 = (lane & 15) * 4;  // tile-load column (floats)

  // Loop-invariant per-lane B-column weights: exactly one of {wc,wf} is 1.0,
  // or w2 = 2.0 for the denominator column; all others zero -> exact select.
  const float wc = (n == 0) ? 1.0f : 0.0f;
  const float wf = (n == 1) ? 1.0f : 0.0f;
  const float w2 = (n == 2) ? 2.0f : 0.0f;

  v8f acc = {0.f, 0.f, 0.f, 0.f, 0.f, 0.f, 0.f, 0.f};

  float4 tv[8];
  float2 cv, fv;
  auto issue_loads = [&](int k0) {
    cv = *(const float2*)(cb + k0 + 2 * lane);
    fv = *(const float2*)(fb + k0 + 2 * lane);
#pragma unroll
    for (int i = 0; i < 8; ++i)
      tv[i] = *(const float4*)(ab + (size_t)(2 * i + rhalf) * TF2_ + k0 + col4);
  };

  issue_loads(0);
  for (int k0 = 0; k0 < TF2_; k0 += KC) {
    // commit staged registers to LDS
    sC[2 * lane] = cv.x; sC[2 * lane + 1] = cv.y;
    sF[2 * lane] = fv.x; sF[2 * lane + 1] = fv.y;
#pragma unroll
    for (int i = 0; i < 8; ++i) {
      float* dst = &sT[(2 * i + rhalf) * TSTR + col4];
      dst[0] = tv[i].x; dst[1] = tv[i].y; dst[2] = tv[i].z; dst[3] = tv[i].w;
    }
    __syncthreads();
    int kn = k0 + KC;
    if (kn < TF2_) issue_loads(kn);   // uniform branch; overlaps with WMMAs below
#pragma unroll
    for (int k4 = 0; k4 < KC; k4 += 4) {
      int ka = k4 + koff;
      v2f a;
      a.x = sT[arow * TSTR + ka];
      a.y = sT[arow * TSTR + ka + 1];
      float c0v = sC[ka], c1v = sC[ka + 1];
      float f0v = sF[ka], f1v = sF[ka + 1];
      v2f bb;
      bb.x = wc * c0v + wf * f0v + w2;
      bb.y = wc * c1v + wf * f1v + w2;
      acc = __builtin_amdgcn_wmma_f32_16x16x4_f32(false, a, false, bb, (short)0, acc,
                                                  false, false);
    }
    __syncthreads();
  }

  // D layout: VGPR v, lanes 0-15: M=v, N=lane; lanes 16-31: M=v+8, N=lane-16.
  // Column 2 holds denom-sum = 2*sum(attn); broadcast it per row via shfl.
#pragma unroll
  for (int v = 0; v < 8; ++v) {
    float av = acc[v];
    float dsum = __shfl(av, (lane & 16) + 2, 32);
    float denom = fmaxf(dsum, 1.0f) + 1e-8f;
    float val = av / denom;
    int row = p_base + (lane >> 4) * 8 + v;
    if (n == 0)      out_curv[(size_t)b * TP_ + row] = val;
    else if (n == 1) out_mean[(size_t)b * TP_ + row] = val;
  }
}

// ---------------------------------------------------------------------------
// Stage 3: masked semitone-deficit energies, z-norm, softmax, ESS.
// One wave32 per batch element; thread 0 does the tiny 16-wide epilogue.
// ---------------------------------------------------------------------------
__global__ __launch_bounds__(512) void energy_kernel(const float* __restrict__ curv_phn,
                                                     const float* __restrict__ mean_phn,
                                                     const float* __restrict__ uv,
                                                     float* __restrict__ out_w,
                                                     float* __restrict__ out_ess) {
  __shared__ float s_e[B_];
  const int tid = threadIdx.x;
  const int b = tid >> 5;
  const int lane = tid & 31;
  const float oodM = 12.0f * log2f(350.0f / 100.0f);
  const float oodK = 12.0f * log2f(1.6f / 100.0f);
  float sM = 0.f, cM = 0.f, sK = 0.f, cK = 0.f;
  for (int p = lane; p < TP_; p += 32) {
    float u = uv[b * TP_ + p];
    float pvM = mean_phn[b * TP_ + p] * u;
    float pvK = curv_phn[b * TP_ + p] * u;
    if (pvM != 0.0f) {
      float semi = 12.0f * log2f(fmaxf(pvM, 1e-6f) * 0.01f);
      sM += fmaxf(oodM - semi, 0.0f);
      cM += 1.0f;
    }
    if (pvK != 0.0f) {
      float semi = 12.0f * log2f(fmaxf(pvK, 1e-6f) * 0.01f);
      sK += fmaxf(oodK - semi, 0.0f);
      cK += 1.0f;
    }
  }
#pragma unroll
  for (int off = 16; off >= 1; off >>= 1) {
    sM += __shfl_xor(sM, off, 32);
    cM += __shfl_xor(cM, off, 32);
    sK += __shfl_xor(sK, off, 32);
    cK += __shfl_xor(cK, off, 32);
  }
  if (lane == 0) s_e[b] = sM / fmaxf(cM, 1.0f) + sK / fmaxf(cK, 1.0f);
  __syncthreads();
  if (tid == 0) {
    float mu = 0.f;
    for (int i = 0; i < B_; ++i) mu += s_e[i];
    mu /= (float)B_;
    float var = 0.f;
    for (int i = 0; i < B_; ++i) { float d = s_e[i] - mu; var += d * d; }
    var /= (float)B_;                       // population std (unbiased=False)
    float sd = fmaxf(sqrtf(var), 1e-6f);
    float zn[B_], mx = -1e30f;
    for (int i = 0; i < B_; ++i) { zn[i] = -(s_e[i] - mu) / sd; mx = fmaxf(mx, zn[i]); }
    float ex[B_], sum = 0.f;
    for (int i = 0; i < B_; ++i) { ex[i] = expf(zn[i] - mx); sum += ex[i]; }
    float s2 = 0.f;
    for (int i = 0; i < B_; ++i) {
      float w = ex[i] / sum;
      out_w[i] = (float)B_ * w;
      s2 += w * w;
    }
    *out_ess = 1.0f / s2;
  }
}

// ---------------------------------------------------------------------------
extern "C" void kernel_launch(void* const* d_in, const int* in_sizes, int n_in,
                              void* d_out, int out_size, void* d_ws, size_t ws_size,
                              hipStream_t stream) {
  (void)in_sizes; (void)n_in; (void)out_size; (void)ws_size;
  const float* f0   = (const float*)d_in[0];   // [B, TF]
  const float* attn = (const float*)d_in[1];   // [B, TP, TF2]
  const float* uv   = (const float*)d_in[2];   // [B, TP]

  float* out       = (float*)d_out;
  float* curv_phn  = out;                       // [B, TP]
  float* mean_phn  = out + (size_t)B_ * TP_;    // [B, TP]
  float* out_w     = out + (size_t)2 * B_ * TP_;       // [B]
  float* out_ess   = out + (size_t)2 * B_ * TP_ + B_;  // [1]

  char* ws = (char*)d_ws;
  int*   prevA  = (int*)(ws);                                   // B*TF ints
  int*   nextA  = (int*)(ws + (size_t)B_ * TF_ * 4);            // B*TF ints
  float* interp = (float*)(ws + (size_t)2 * B_ * TF_ * 4);      // B*TF floats
  float* curv2  = (float*)(ws + (size_t)3 * B_ * TF_ * 4);      // B*TF2 floats
  float* f02    = (float*)(ws + (size_t)3 * B_ * TF_ * 4 + (size_t)B_ * TF2_ * 4);

  scan_kernel<<<B_, 32, 0, stream>>>(f0, prevA, nextA);
  interp_kernel<<<(B_ * TF_) / 256, 256, 0, stream>>>(f0, prevA, nextA, interp);
  pairsum_kernel<<<(B_ * TF2_) / 256, 256, 0, stream>>>(f0, interp, curv2, f02);
  einsum_wmma_kernel<<<dim3(TP_ / 16, B_), 32, 0, stream>>>(attn, curv2, f02,
                                                            curv_phn, mean_phn);
  energy_kernel<<<1, 512, 0, stream>>>(curv_phn, mean_phn, uv, out_w, out_ess);
}